// MVTracker_52527450030080
// MI455X (gfx1250) — compile-verified
//
#include <hip/hip_runtime.h>

// MVTracker fused KNN + grouped-correlation kernel for MI455X (gfx1250).
// Shapes fixed by setup_inputs(): B=4, N=16384, M=2048, C=128, k=16, groups=4.

typedef __attribute__((ext_vector_type(2))) float v2f;
typedef __attribute__((ext_vector_type(8))) float v8f;

#define B_ 4
#define N_ 16384
#define M_ 2048
#define C_ 128
#define K_ 16
#define CHUNK 1024            // points staged to LDS per outer iteration
#define NCHUNK (N_ / CHUNK)   // 16
#define WAVES 4
#define TPB (WAVES * 32)      // 128 threads = 4 wave32

__global__ __launch_bounds__(TPB) void mvtracker_fused(
    const float* __restrict__ xyz,       // [B,N,3]
    const float* __restrict__ fvec,      // [B,N,C]
    const float* __restrict__ targets,   // [B,M,C]
    const float* __restrict__ cw,        // [B,M,3] query coords
    float* __restrict__ out)             // [B,M,K,10]
{
  __shared__ __align__(16) float pts[CHUNK * 3];  // staged xyz chunk (async target)
  __shared__ float xx[CHUNK];            // ||x||^2 per staged point
  __shared__ float bd[WAVES * 16 * 16];  // per-wave top-16 distances per query
  __shared__ int   bi[WAVES * 16 * 16];  // per-wave top-16 indices
  __shared__ float thr[WAVES * 16];      // per-wave kth-best threshold per query
  __shared__ float qpt[16][3];           // query coords
  __shared__ float qq[16];               // ||q||^2
  __shared__ int   si[16 * K_];          // merged + sorted neighbor indices
  __shared__ float tl[C_];               // target vector of current query

  const int tid  = threadIdx.x;
  const int lane = tid & 31;
  const int w    = tid >> 5;
  const int b    = blockIdx.x >> 7;           // 128 m-tiles per batch
  const int m0   = (blockIdx.x & 127) * 16;

  // ---------------- init ----------------
  if (tid < 16) {
    const float* qp = cw + ((size_t)b * M_ + m0 + tid) * 3;
    float x = qp[0], y = qp[1], z = qp[2];
    qpt[tid][0] = x; qpt[tid][1] = y; qpt[tid][2] = z;
    qq[tid] = x * x + y * y + z * z;
  }
  for (int j = tid; j < WAVES * 16 * 16; j += TPB) { bd[j] = 3.4e38f; bi[j] = 0x7fffffff; }
  if (tid < WAVES * 16) thr[tid] = 3.4e38f;
  __syncthreads();

  // Lane-half selectors (branchless fragment construction):
  // lanes 0-15 carry (x,y) = dims K0/K1, lanes 16-31 carry (z,0) = dims K2/K3.
  const int   lo16  = (lane < 16);
  const int   xsel  = lo16 ? 0 : 2;      // x vs z element within a point
  const float ymask = lo16 ? 1.0f : 0.0f;
  const int   half8 = lo16 ? 0 : 8;

  // A fragment: 16 query rows x K=4 (x,y,z,0), per ISA 32-bit A 16x4 layout.
  v2f afrag;
  {
    int row = lane & 15;
    afrag.x = qpt[row][xsel];
    afrag.y = qpt[row][1] * ymask;
  }

  // register caches: ||q||^2 per row this lane touches, and threshold cache
  float qqr[8];
#pragma unroll
  for (int r = 0; r < 8; ++r) qqr[r] = qq[r + half8];
  float thrc[8];
#pragma unroll
  for (int r = 0; r < 8; ++r) thrc[r] = 3.4e38f;

  // ---------------- phase 1: fused distance tiles (WMMA) + top-k ----------------
  for (int ch = 0; ch < NCHUNK; ++ch) {
    __syncthreads();
    const float* src = xyz + ((size_t)b * N_ + (size_t)ch * CHUNK) * 3;
    if (ch + 1 < NCHUNK) __builtin_prefetch(src + CHUNK * 3, 0, 0);  // global_prefetch_b8

    // async-copy the 12KB xyz chunk into LDS as float4s (ASYNCcnt path)
#pragma unroll
    for (int j = 0; j < (CHUNK * 3) / (TPB * 4); ++j) {   // 6 x b128 per thread
      const int e4 = tid + TPB * j;                       // float4 index
      unsigned ldsoff = (unsigned)(size_t)&pts[e4 * 4];   // low 32b of generic = LDS offset
      const float* g = src + (size_t)e4 * 4;
      asm volatile("global_load_async_to_lds_b128 %0, %1, off"
                   :: "v"(ldsoff), "v"(g) : "memory");
    }
    asm volatile("s_wait_asynccnt 0" ::: "memory");
    __syncthreads();

#pragma unroll
    for (int j = 0; j < CHUNK / TPB; ++j) {
      int p = tid + TPB * j;
      float x = pts[3 * p], y = pts[3 * p + 1], z = pts[3 * p + 2];
      xx[p] = x * x + y * y + z * z;
    }
    __syncthreads();

    // each wave handles its quarter of the chunk: 16 tiles of 16 points
    for (int t = 0; t < (CHUNK / WAVES) / 16; ++t) {
      const int lp = w * (CHUNK / WAVES) + t * 16 + (lane & 15);

      // Branchless B fragment: both LDS loads unconditional, half-select folded
      // into the address (x vs z) and a multiplicative mask (y vs 0).
      v2f bfrag;
      bfrag.x = pts[3 * lp + xsel];
      bfrag.y = pts[3 * lp + 1] * ymask;

      v8f c = {};
      c = __builtin_amdgcn_wmma_f32_16x16x4_f32(
          /*neg_a=*/false, afrag, /*neg_b=*/false, bfrag,
          /*c_mod=*/(short)0, c, /*reuse_a=*/false, /*reuse_b=*/false);

      const float xxl = xx[lp];
      const int   n_g = ch * CHUNK + lp;  // global point index of this lane's column
#pragma unroll
      for (int r = 0; r < 8; ++r) {
        // C/D layout: VGPR r holds row r (lanes 0-15) and row r+8 (lanes 16-31)
        const int tq = w * 16 + r + half8;
        float d2   = qqr[r] + xxl - 2.0f * c[r];
        bool  cand = d2 < thrc[r];
        unsigned mb = __builtin_amdgcn_ballot_w32(cand);
        if (mb) {
          do {  // serialize rare insertions; LDS ops from one wave are in-order
            int srcl = __builtin_ctz(mb);
            mb &= mb - 1;
            if (lane == srcl) {
              const int base = tq * 16;
              float mx = -1.0f; int ms = 0;
#pragma unroll
              for (int jj = 0; jj < 16; ++jj) {
                float v = bd[base + jj];
                if (v > mx) { mx = v; ms = jj; }
              }
              if (d2 < mx) {
                bd[base + ms] = d2;
                bi[base + ms] = n_g;
                float nm = -1.0f;
#pragma unroll
                for (int jj = 0; jj < 16; ++jj) {
                  float v = bd[base + jj];
                  if (v > nm) nm = v;
                }
                thr[tq] = nm;
              }
            }
          } while (mb);
          thrc[r] = thr[tq];  // refresh cached threshold after any insertion
        }
      }
    }
  }

  // ---------------- phase 2: merge 4 per-wave lists -> sorted top-16 ----------------
  __syncthreads();
  for (int pr = tid; pr < 16 * 64; pr += TPB) {
    int   q = pr >> 6, s = pr & 63;
    int   e = ((s >> 4) * 16 + q) * 16 + (s & 15);
    float d = bd[e];
    int   i = bi[e];
    int rank = 0;
    for (int s2 = 0; s2 < 64; ++s2) {
      int   e2  = ((s2 >> 4) * 16 + q) * 16 + (s2 & 15);
      float d2v = bd[e2];
      int   i2  = bi[e2];
      rank += (d2v < d) || ((d2v == d) && (i2 < i));  // index tie-break = top_k order
    }
    if (rank < K_) si[q * K_ + rank] = i;
  }
  __syncthreads();

  // ---------------- phase 3: gather + grouped correlation + emit ----------------
  const float inv_sqrt_cg = 0.17677669529663687f;  // 1/sqrt(32)
  for (int q = 0; q < 16; ++q) {
    __syncthreads();
    tl[tid] = targets[((size_t)b * M_ + m0 + q) * C_ + tid];
    __syncthreads();

    const int kk   = tid >> 3;   // neighbor 0..15
    const int part = tid & 7;    // 16-channel slice; group g = part/2
    const int nidx = si[q * K_ + kk];

    const float* frow = fvec + ((size_t)b * N_ + nidx) * C_ + part * 16;
    float acc = 0.0f;
#pragma unroll
    for (int cc = 0; cc < 16; ++cc) acc += frow[cc] * tl[part * 16 + cc];
    acc += __shfl_xor(acc, 1);   // combine the two 16-ch halves of each group

    const size_t ob = ((size_t)(b * M_ + m0 + q) * K_ + kk) * 10;
    if ((part & 1) == 0) out[ob + (part >> 1)] = acc * inv_sqrt_cg;  // corrs[0..3]
    if (part == 1 || part == 3) {
      const float* np = xyz + ((size_t)b * N_ + nidx) * 3;
      float nx = np[0], ny = np[1], nz = np[2];
      if (part == 1) {           // offset = neighbor_xyz - query_xyz
        out[ob + 4] = nx - qpt[q][0];
        out[ob + 5] = ny - qpt[q][1];
        out[ob + 6] = nz - qpt[q][2];
      } else {                   // raw neighbor_xyz
        out[ob + 7] = nx;
        out[ob + 8] = ny;
        out[ob + 9] = nz;
      }
    }
  }
}

extern "C" void kernel_launch(void* const* d_in, const int* in_sizes, int n_in,
                              void* d_out, int out_size, void* d_ws, size_t ws_size,
                              hipStream_t stream) {
  (void)in_sizes; (void)n_in; (void)d_ws; (void)ws_size; (void)out_size;
  const float* xyz     = (const float*)d_in[0];
  const float* fvec    = (const float*)d_in[1];
  const float* targets = (const float*)d_in[2];
  const float* cw      = (const float*)d_in[3];
  float* out = (float*)d_out;
  dim3 grid(B_ * (M_ / 16));   // 512 blocks, one 16-query tile each
  mvtracker_fused<<<grid, TPB, 0, stream>>>(xyz, fvec, targets, cw, out);
}